// DiagonalwiseRefactorization_67242007987146
// MI455X (gfx1250) — compile-verified
//
#include <hip/hip_runtime.h>
#include <stdint.h>

#define NBATCH 16
#define CHN    512
#define HDIM   64
#define WDIM   64
#define OH     63
#define OW     63
#define PLANE  (HDIM*WDIM)     // 4096 elements per input plane
#define OPLANE (OH*OW)         // 3969 elements per output plane
#define TS     66              // LDS row stride: 64 data + 2 pad (TDM pad feature)
#define LDS_ELEMS (66*66 + 68) // 4424: covers max read index 4423 (overrun reads, zeroed)

typedef uint32_t v4u __attribute__((ext_vector_type(4)));
typedef int      v8i __attribute__((ext_vector_type(8)));
typedef int      v4i __attribute__((ext_vector_type(4)));
typedef float    v2f __attribute__((ext_vector_type(2)));
typedef float    v8f __attribute__((ext_vector_type(8)));

// Depthwise 4x4 conv, stride 1, pad 1, computed as banded-Toeplitz GEMM on WMMA.
// LDS layout: element (iy, ix) at P = 1 + (iy+1)*TS + ix; per-row 2-DWORD pads
// (written as zeros) double as the ix=-1 / ix=64 ghost columns. Tap (iy',ix')
// reads P = (iy'+1)*TS + ix' + 1.
__global__ __launch_bounds__(256) void dwconv4x4_wmma_kernel(
    const float* __restrict__ x,
    const float* __restrict__ weight,
    const float* __restrict__ mask,
    float* __restrict__ out)
{
    __shared__ float tile[LDS_ELEMS];
    __shared__ float Btab[4 * 20 * 16];   // banded weights: Btab[ky][j][n] = W[ky, j-n]

    const int nc  = blockIdx.x;        // n*CHN + c
    const int c   = nc & (CHN - 1);
    const int tid = threadIdx.x;

    const float* xp = x + (size_t)nc * PLANE;

    // ---- per-channel 4x4 taps (masked) — uniform across block -> scalar loads ----
    const int ci = c & 63;
    const float* wp = weight + ((size_t)c * 64 + ci) * 16;
    const float* mp = mask   + ((size_t)c * 64 + ci) * 16;
    float wt[16];
    #pragma unroll
    for (int j = 0; j < 16; ++j) wt[j] = wp[j] * mp[j];

    // ---- zero halo cells (disjoint from TDM-written interior) ----
    if (tid < 67) {
        tile[tid]               = 0.0f;   // ghost row iy=-1 : P 0..66
        tile[65*TS + tid]       = 0.0f;   // P 4290..4356 (ghost row iy=64)
        tile[65*TS + 67 + tid]  = 0.0f;   // P 4357..4423 (overrun region for discarded lanes)
    }
    if (tid < 64) {
        tile[1 + (tid+1)*TS + 64] = 0.0f; // right pad slot (= ghost ix=64)
        tile[1 + (tid+1)*TS + 65] = 0.0f; // second pad slot (= ghost ix=-1 of next row)
    }

#if __has_builtin(__builtin_amdgcn_tensor_load_to_lds)
    if (tid < 32) {  // wave 0 issues one TDM op for the whole plane
        uint32_t lds_off = (uint32_t)(uintptr_t)(&tile[67]); // generic LDS addr low bits = LDS byte offset
        uint64_t ga      = (uint64_t)(uintptr_t)xp;

        v4u g0;
        g0.x = 1u;                                    // count=1, is_restore=0, gather=0
        g0.y = lds_off;                               // lds_addr (bytes)
        g0.z = (uint32_t)(ga & 0xffffffffu);          // global_addr[31:0]
        g0.w = (uint32_t)((ga >> 32) & 0x01ffffffu)   // global_addr[56:32]
             | (2u << 30);                            // type = 2 ("image")

        v8i g1;
        g1[0] = (int)((2u << 16)                      // data_size = 4 bytes
                    | (1u << 20)                      // pad_enable
                    | (5u << 22)                      // pad_interval: 2^(5+1) = 64 DWORDs
                    | (1u << 25));                    // pad_amount: 2 DWORDs
        g1[1] = (int)(64u << 16);                     // tensor_dim0[15:0] = 64
        g1[2] = (int)(64u << 16);                     // tensor_dim1[15:0] = 64
        g1[3] = (int)(64u << 16);                     // tile_dim0 = 64
        g1[4] = (int)(64u);                           // tile_dim1 = 64, tile_dim2 = 0
        g1[5] = 64;                                   // tensor_dim0_stride = 64
        g1[6] = 0;
        g1[7] = 0;

        v4i g2 = {0, 0, 0, 0};                        // 2D tile: groups 2/3 unused
        v4i g3 = {0, 0, 0, 0};

#if __has_include(<hip/amd_detail/amd_gfx1250_TDM.h>)
        v8i gz = {0, 0, 0, 0, 0, 0, 0, 0};
        __builtin_amdgcn_tensor_load_to_lds(g0, g1, g2, g3, gz, 0);
#else
        __builtin_amdgcn_tensor_load_to_lds(g0, g1, g2, g3, 0);
#endif
#if __has_builtin(__builtin_amdgcn_s_wait_tensorcnt)
        __builtin_amdgcn_s_wait_tensorcnt((short)0);
#else
        asm volatile("s_wait_tensorcnt 0" ::: "memory");
#endif
    }
#else
    // Fallback: cooperative direct copy into the padded layout
    for (int i = tid; i < PLANE; i += 256) {
        int iy = i >> 6, ix = i & 63;
        tile[1 + (iy + 1)*TS + ix] = xp[i];
    }
#endif

    // ---- build banded-B table once per block (ky is compile-time -> no dyn reg index) ----
    #pragma unroll
    for (int ky = 0; ky < 4; ++ky) {
        const float w0 = wt[ky*4+0], w1 = wt[ky*4+1], w2 = wt[ky*4+2], w3 = wt[ky*4+3];
        #pragma unroll
        for (int r = 0; r < 2; ++r) {
            const int e = r*256 + tid;            // 0..511; need e < 320
            if (e < 320) {
                const int n = e & 15;
                const int j = e >> 4;             // 0..19
                const int d = j - n;
                Btab[ky*320 + e] =
                    (d==0)?w0:((d==1)?w1:((d==2)?w2:((d==3)?w3:0.f)));
            }
        }
    }

    __syncthreads();

    // ---- WMMA lane roles (wave32) ----
    const int lane  = tid & 31;
    const int wave  = tid >> 5;           // 0..7
    const int m     = lane & 15;          // A: row M; B: col N; D: col N
    const int khalf = (lane >> 4) << 1;   // K pair base: 0 (lanes 0-15) / 2 (lanes 16-31)

    // ---- load per-lane B fragments from the table (pairs 64B apart -> ds_load_2addr) ----
    v2f Bm[4][5];
    #pragma unroll
    for (int ky = 0; ky < 4; ++ky) {
        #pragma unroll
        for (int c2 = 0; c2 < 5; ++c2) {
            const int j0 = 4*c2 + khalf;
            const float* bp = &Btab[ky*320 + j0*16 + m];
            v2f b;
            b.x = bp[0];
            b.y = bp[16];
            Bm[ky][c2] = b;
        }
    }

    // ---- 16 output tiles of 16x16; 2 per wave; 20 chained K=4 WMMAs per tile ----
    float* op = out + (size_t)nc * OPLANE;
    #pragma unroll
    for (int i = 0; i < 2; ++i) {
        const int tix = wave*2 + i;
        const int oy0 = (tix >> 2) * 16;
        const int ox0 = (tix & 3) * 16;

        v8f acc = {0.f,0.f,0.f,0.f,0.f,0.f,0.f,0.f};
        #pragma unroll
        for (int ky = 0; ky < 4; ++ky) {
            #pragma unroll
            for (int c2 = 0; c2 < 5; ++c2) {
                // A[m, k] = X[oy0+m+ky-1, ox0+(4*c2+k)-1] from padded LDS
                const int P = (oy0 + m + ky)*TS + ox0 + 4*c2 + khalf;
                v2f a;
                a.x = tile[P];
                a.y = tile[P + 1];
                acc = __builtin_amdgcn_wmma_f32_16x16x4_f32(
                        false, a, false, Bm[ky][c2], (short)0, acc, false, false);
            }
        }

        // D layout: VGPR v -> M = v + 8*(lane>=16), N = lane&15
        const int ox    = ox0 + m;
        const int mbase = oy0 + ((lane >> 4) << 3);
        #pragma unroll
        for (int v = 0; v < 8; ++v) {
            const int oy = mbase + v;
            if (oy < OH && ox < OW)
                __builtin_nontemporal_store(acc[v], op + oy*OW + ox);
        }
    }
}

extern "C" void kernel_launch(void* const* d_in, const int* in_sizes, int n_in,
                              void* d_out, int out_size, void* d_ws, size_t ws_size,
                              hipStream_t stream) {
    (void)in_sizes; (void)n_in; (void)out_size; (void)d_ws; (void)ws_size;
    const float* x      = (const float*)d_in[0];
    const float* weight = (const float*)d_in[1];
    const float* mask   = (const float*)d_in[2];
    // d_in[3] = groups (8), d_in[4] = stride (1): fixed by the reference shapes.
    float* out = (float*)d_out;

    dwconv4x4_wmma_kernel<<<NBATCH * CHN, 256, 0, stream>>>(x, weight, mask, out);
}